// DA_RNN_Explainable_73504070304220
// MI455X (gfx1250) — compile-verified
//
#include <hip/hip_runtime.h>
#include <math.h>

// DA-RNN on MI455X (gfx1250): persistent kernel, B split across 8 workgroups,
// h kept in LDS, matmuls via V_WMMA_F32_16X16X4_F32, x_t staged with
// double-buffered GLOBAL_LOAD_ASYNC_TO_LDS_B128 (ASYNCcnt pipeline depth 2),
// temporal attention via online softmax (t-scores parked in the beta output).
// tanh on the recurrent critical path uses hardware V_TANH_F32 when available.

#define B_  128
#define T_  256
#define D_  256
#define H_  512
#define BT  16            // batch rows per workgroup

#define XS_S (D_ + 2)     // LDS row strides (+2 pad: dodge bank conflicts on A reads)
#define HS_S (H_ + 2)
#define SC_S (D_ + 2)
#define RZ_S (2 * H_)     // r|z gates; cols [0,512) later overwritten with h_new

typedef __attribute__((ext_vector_type(2))) float v2f;
typedef __attribute__((ext_vector_type(8))) float v8f;

__device__ __forceinline__ v8f wmma4(v2f a, v2f b, v8f c) {
  // D = A(16x4 f32) * B(4x16 f32) + C(16x16 f32), wave32
  return __builtin_amdgcn_wmma_f32_16x16x4_f32(false, a, false, b, (short)0, c,
                                               false, false);
}

__device__ __forceinline__ float fast_tanh(float v) {
#if __has_builtin(__builtin_amdgcn_tanhf)
  return __builtin_amdgcn_tanhf(v);        // V_TANH_F32 (gfx1250 TRANS op)
#else
  return tanhf(v);
#endif
}

// Async 16B copy global -> LDS (per active lane). Generic LDS pointers carry
// the LDS byte offset in their low 32 bits.
__device__ __forceinline__ void async_ld_b128(const float* gptr, float* lptr) {
  const unsigned int loff = (unsigned int)(unsigned long long)lptr;
  asm volatile("global_load_async_to_lds_b128 %0, %1, off"
               :: "v"(loff), "v"(gptr) : "memory");
}
__device__ __forceinline__ void wait_async_le2() {
  asm volatile("s_wait_asynccnt 0x2" ::: "memory");
}
__device__ __forceinline__ void wait_async_0() {
  asm volatile("s_wait_asynccnt 0x0" ::: "memory");
}

__global__ __launch_bounds__(512)
void darnn_kernel(const float* __restrict__ x,
                  const float* __restrict__ Wa, const float* __restrict__ ba,
                  const float* __restrict__ Wi, const float* __restrict__ bi,
                  const float* __restrict__ Wh, const float* __restrict__ bh,
                  const float* __restrict__ Wt, const float* __restrict__ bt,
                  const float* __restrict__ Wf, const float* __restrict__ bf,
                  float* __restrict__ out0, float* __restrict__ outA,
                  float* __restrict__ outB)
{
  __shared__ float xs[2][BT * XS_S]; // double-buffered x_t tile (async target)
  __shared__ float hs[BT * HS_S];    // hidden state (f32)
  __shared__ float sc[BT * SC_S];    // attn scores -> wx (in place)
  __shared__ float rz[BT * RZ_S];    // sigmoid(r)|sigmoid(z); cols[0,512)->h_new

  const int tid  = threadIdx.x;
  const int wave = tid >> 5;
  const int lane = tid & 31;
  const int half = lane >> 4;
  const int lm   = lane & 15;
  const int b0   = blockIdx.x * BT;

  // each thread stages 2 x 16B chunks of the 16x256 x-tile
  const int srow0 = tid >> 6,          sc40 = (tid & 63) << 2;
  const int srow1 = (tid + 512) >> 6,  sc41 = ((tid + 512) & 63) << 2;

  for (int i = tid; i < BT * HS_S; i += 512) hs[i] = 0.f;

  // kick off async staging of x_0 into buffer 0
  async_ld_b128(&x[((size_t)(b0 + srow0) * T_ + 0) * D_ + sc40],
                &xs[0][srow0 * XS_S + sc40]);
  async_ld_b128(&x[((size_t)(b0 + srow1) * T_ + 0) * D_ + sc41],
                &xs[0][srow1 * XS_S + sc41]);

  float cacc[16];
  #pragma unroll
  for (int i = 0; i < 16; ++i) cacc[i] = 0.f;
  float mrun = -INFINITY, srun = 0.f;

  for (int t = 0; t < T_; ++t) {
    const int cur = t & 1;
    // ---- phase 1: async-issue x_{t+1}, wait for x_t's pair, sync ----
    if (t + 1 < T_) {
      const int nxt = cur ^ 1;
      async_ld_b128(&x[((size_t)(b0 + srow0) * T_ + (t + 1)) * D_ + sc40],
                    &xs[nxt][srow0 * XS_S + sc40]);
      async_ld_b128(&x[((size_t)(b0 + srow1) * T_ + (t + 1)) * D_ + sc41],
                    &xs[nxt][srow1 * XS_S + sc41]);
      wait_async_le2();   // older pair (x_t) complete; x_{t+1} may be in flight
    } else {
      wait_async_0();
    }
    __syncthreads();
    const float* xcur = xs[cur];

    // ---- phase 2: scores = tanh([x_t, h] @ W_attn + b_attn); wave -> 16 cols ----
    {
      const int n0 = wave * 16;
      const float bias = ba[n0 + lm];
      v8f acc;
      #pragma unroll
      for (int r = 0; r < 8; ++r) acc[r] = bias;
      #pragma unroll 2
      for (int kb = 0; kb < D_; kb += 4) {                 // x part, K=256
        v2f a, bm;
        const float* ap = &xcur[lm * XS_S + kb + 2 * half];
        a.x = ap[0]; a.y = ap[1];
        const float* bp = &Wa[(size_t)(kb + 2 * half) * D_ + n0 + lm];
        bm.x = bp[0]; bm.y = bp[D_];
        acc = wmma4(a, bm, acc);
      }
      #pragma unroll 2
      for (int kb = 0; kb < H_; kb += 4) {                 // h part, K=512
        v2f a, bm;
        const float* ap = &hs[lm * HS_S + kb + 2 * half];
        a.x = ap[0]; a.y = ap[1];
        const float* bp = &Wa[(size_t)(D_ + kb + 2 * half) * D_ + n0 + lm];
        bm.x = bp[0]; bm.y = bp[D_];
        acc = wmma4(a, bm, acc);
      }
      #pragma unroll
      for (int r = 0; r < 8; ++r)
        sc[(r + 8 * half) * SC_S + n0 + lm] = fast_tanh(acc[r]);
    }
    __syncthreads();

    // ---- phase 3: softmax over D per row; alpha -> global; wx -> sc ----
    {
      const int row = wave;
      float vals[8], mx = -INFINITY;
      #pragma unroll
      for (int i = 0; i < 8; ++i) {
        vals[i] = sc[row * SC_S + lane + 32 * i];
        mx = fmaxf(mx, vals[i]);
      }
      #pragma unroll
      for (int o = 16; o > 0; o >>= 1) mx = fmaxf(mx, __shfl_xor(mx, o, 32));
      float sum = 0.f;
      #pragma unroll
      for (int i = 0; i < 8; ++i) { vals[i] = __expf(vals[i] - mx); sum += vals[i]; }
      #pragma unroll
      for (int o = 16; o > 0; o >>= 1) sum += __shfl_xor(sum, o, 32);
      const float inv = 1.f / sum;
      float* arow = &outA[((size_t)(b0 + row) * T_ + t) * D_];
      #pragma unroll
      for (int i = 0; i < 8; ++i) {
        const int d = lane + 32 * i;
        const float al = vals[i] * inv;
        arow[d] = al;
        sc[row * SC_S + d] = al * xcur[row * XS_S + d];
      }
    }
    __syncthreads();

    // ---- phase 4: r,z gates (N=0..1023): sigmoid(wx@Wi^T + h@Wh^T + bi+bh) ----
    #pragma unroll 1
    for (int tt = 0; tt < 4; ++tt) {
      const int n = (wave * 4 + tt) * 16 + lm;             // gate col 0..1023
      const float bias = bi[n] + bh[n];
      v8f acc;
      #pragma unroll
      for (int r = 0; r < 8; ++r) acc[r] = bias;
      #pragma unroll 2
      for (int kb = 0; kb < D_; kb += 4) {                 // wx @ Wi^T
        v2f a, bm;
        const float* ap = &sc[lm * SC_S + kb + 2 * half];
        a.x = ap[0]; a.y = ap[1];
        const float* bp = &Wi[(size_t)n * D_ + kb + 2 * half];
        bm.x = bp[0]; bm.y = bp[1];
        acc = wmma4(a, bm, acc);
      }
      #pragma unroll 2
      for (int kb = 0; kb < H_; kb += 4) {                 // h @ Wh^T
        v2f a, bm;
        const float* ap = &hs[lm * HS_S + kb + 2 * half];
        a.x = ap[0]; a.y = ap[1];
        const float* bp = &Wh[(size_t)n * H_ + kb + 2 * half];
        bm.x = bp[0]; bm.y = bp[1];
        acc = wmma4(a, bm, acc);
      }
      #pragma unroll
      for (int r = 0; r < 8; ++r)
        rz[(r + 8 * half) * RZ_S + n] = 1.f / (1.f + __expf(-acc[r]));
    }
    __syncthreads();

    // ---- phase 5: n gate + h update; h_new overwrites rz cols [0,512) ----
    #pragma unroll 1
    for (int tt = 0; tt < 2; ++tt) {
      const int j = (wave * 2 + tt) * 16 + lm;             // hidden col 0..511
      const int n = 1024 + j;                              // weight row
      v8f ai, ah;
      { const float bii = bi[n], bhh = bh[n];
        #pragma unroll
        for (int r = 0; r < 8; ++r) { ai[r] = bii; ah[r] = bhh; } }
      #pragma unroll 2
      for (int kb = 0; kb < D_; kb += 4) {                 // gi_n
        v2f a, bm;
        const float* ap = &sc[lm * SC_S + kb + 2 * half];
        a.x = ap[0]; a.y = ap[1];
        const float* bp = &Wi[(size_t)n * D_ + kb + 2 * half];
        bm.x = bp[0]; bm.y = bp[1];
        ai = wmma4(a, bm, ai);
      }
      #pragma unroll 2
      for (int kb = 0; kb < H_; kb += 4) {                 // gh_n
        v2f a, bm;
        const float* ap = &hs[lm * HS_S + kb + 2 * half];
        a.x = ap[0]; a.y = ap[1];
        const float* bp = &Wh[(size_t)n * H_ + kb + 2 * half];
        bm.x = bp[0]; bm.y = bp[1];
        ah = wmma4(a, bm, ah);
      }
      float hnew[8];
      #pragma unroll
      for (int r = 0; r < 8; ++r) {
        const int m = r + 8 * half;
        const float rv = rz[m * RZ_S + j];
        const float zv = rz[m * RZ_S + 512 + j];
        const float na = fast_tanh(ai[r] + rv * ah[r]);
        hnew[r] = (1.f - zv) * na + zv * hs[m * HS_S + j];
      }
      #pragma unroll
      for (int r = 0; r < 8; ++r) rz[(r + 8 * half) * RZ_S + j] = hnew[r];
    }
    __syncthreads();

    // ---- phase 6: commit h, temporal score + online softmax/context ----
    {
      const int row = wave;                                // wave <-> batch row
      float td = 0.f, hn[16];
      #pragma unroll
      for (int i = 0; i < 16; ++i) {
        const int jj = lane + 32 * i;
        hn[i] = rz[row * RZ_S + jj];
        hs[row * HS_S + jj] = hn[i];
        td += hn[i] * Wt[jj];
      }
      #pragma unroll
      for (int o = 16; o > 0; o >>= 1) td += __shfl_xor(td, o, 32);
      const float score = fast_tanh(td + bt[0]);
      const float mnew  = fmaxf(mrun, score);
      const float corr  = __expf(mrun - mnew);
      const float p     = __expf(score - mnew);
      srun = srun * corr + p;
      #pragma unroll
      for (int i = 0; i < 16; ++i) cacc[i] = cacc[i] * corr + p * hn[i];
      mrun = mnew;
      if (lane == 0) outB[(size_t)(b0 + row) * T_ + t] = score; // park t-score
    }
    // next iteration's barrier (after async wait) orders hs vs readers
  }
  __syncthreads();

  // ---- epilogue: beta (from parked t-scores), context, logits ----
  {
    const int row = wave;
    const int gb  = b0 + row;
    const float invs = 1.f / srun;
    float* brow = &outB[(size_t)gb * T_];
    #pragma unroll
    for (int i = 0; i < 8; ++i) {
      const int tt = lane + 32 * i;
      const float scv = brow[tt];                 // same-wave store->load: ordered
      brow[tt] = __expf(scv - mrun) * invs;
    }
    float cd = 0.f;
    #pragma unroll
    for (int i = 0; i < 16; ++i) {
      const int jj = lane + 32 * i;
      cd += (cacc[i] * invs) * Wf[jj];
    }
    #pragma unroll
    for (int o = 16; o > 0; o >>= 1) cd += __shfl_xor(cd, o, 32);
    if (lane == 0) out0[gb] = 1.f / (1.f + __expf(-(cd + bf[0])));
  }
}

extern "C" void kernel_launch(void* const* d_in, const int* in_sizes, int n_in,
                              void* d_out, int out_size, void* d_ws, size_t ws_size,
                              hipStream_t stream) {
  (void)in_sizes; (void)n_in; (void)out_size; (void)d_ws; (void)ws_size;
  const float* x  = (const float*)d_in[0];
  const float* Wa = (const float*)d_in[1];
  const float* ba = (const float*)d_in[2];
  const float* Wi = (const float*)d_in[3];
  const float* bi = (const float*)d_in[4];
  const float* Wh = (const float*)d_in[5];
  const float* bh = (const float*)d_in[6];
  const float* Wt = (const float*)d_in[7];
  const float* bt = (const float*)d_in[8];
  const float* Wf = (const float*)d_in[9];
  const float* bf = (const float*)d_in[10];

  float* out0 = (float*)d_out;                                  // [B,1]
  float* outA = (float*)d_out + B_;                             // [B,T,D]
  float* outB = (float*)d_out + B_ + (size_t)B_ * T_ * D_;      // [B,T,1]

  darnn_kernel<<<dim3(B_ / BT), dim3(512), 0, stream>>>(
      x, Wa, ba, Wi, bi, Wh, bh, Wt, bt, Wf, bf, out0, outA, outB);
}